// MLP_67748814127321
// MI455X (gfx1250) — compile-verified
//
#include <hip/hip_runtime.h>

// Problem sizes (fixed by the reference)
#define B_  64
#define S_  1024   // IN_LEN
#define NV  512    // N_VARS
#define P_  1024   // PRED_LEN
#define E_  8      // experts
// TOPK = 2

typedef __attribute__((ext_vector_type(2))) float v2f;
typedef __attribute__((ext_vector_type(8))) float v8f;

// ---------------------------------------------------------------------------
// Kernel 1: gating. One thread per token (b, n).
//   logits[e] = sum_s x[b,s,n] * gate_W[s,e]   (x reads coalesced across n)
//   gate = softmax(logits); top-2 scatter -> combine[b,n,e]; also store gate.
// ---------------------------------------------------------------------------
__global__ __launch_bounds__(256) void gate_kernel(
    const float* __restrict__ x,       // [B,S,NV]
    const float* __restrict__ gate_W,  // [S,E]
    float* __restrict__ combine,       // ws [B,NV,E]
    float* __restrict__ gate_full)     // ws [B,NV,E]
{
  const int token = blockIdx.x * blockDim.x + threadIdx.x;  // 0..B*NV-1
  const int b = token >> 9;          // /NV (=512)
  const int n = token & (NV - 1);

  float acc[E_];
#pragma unroll
  for (int e = 0; e < E_; ++e) acc[e] = 0.0f;

  const float* xp = x + (size_t)b * S_ * NV + n;
  for (int s = 0; s < S_; ++s) {
    const float xv = xp[(size_t)s * NV];
    const float* gw = gate_W + s * E_;   // uniform across wave -> s_loads
#pragma unroll
    for (int e = 0; e < E_; ++e) acc[e] = fmaf(xv, gw[e], acc[e]);
  }

  // softmax over 8 logits
  float m = acc[0];
#pragma unroll
  for (int e = 1; e < E_; ++e) m = fmaxf(m, acc[e]);
  float g[E_];
  float sum = 0.0f;
#pragma unroll
  for (int e = 0; e < E_; ++e) { g[e] = __expf(acc[e] - m); sum += g[e]; }
  const float inv = 1.0f / sum;
#pragma unroll
  for (int e = 0; e < E_; ++e) g[e] *= inv;

  // top-2 (first-index tiebreak like lax.top_k)
  int i1 = 0; float v1 = g[0];
#pragma unroll
  for (int e = 1; e < E_; ++e) { if (g[e] > v1) { v1 = g[e]; i1 = e; } }
  int i2 = -1; float v2 = -1.0f;
#pragma unroll
  for (int e = 0; e < E_; ++e) { if (e != i1 && g[e] > v2) { v2 = g[e]; i2 = e; } }

  float* cp = combine   + (size_t)token * E_;
  float* gp = gate_full + (size_t)token * E_;
#pragma unroll
  for (int e = 0; e < E_; ++e) {
    cp[e] = (e == i1) ? v1 : ((e == i2) ? v2 : 0.0f);
    gp[e] = g[e];
  }
}

// ---------------------------------------------------------------------------
// Kernel 2: deterministic gate_mean[n,e] = mean_b gate[b,n,e]
// ---------------------------------------------------------------------------
__global__ __launch_bounds__(256) void gate_mean_kernel(
    const float* __restrict__ gate_full,  // [B,NV,E]
    float* __restrict__ gm)               // [NV,E]
{
  const int idx = blockIdx.x * blockDim.x + threadIdx.x;  // 0..NV*E-1
  float s = 0.0f;
  for (int b = 0; b < B_; ++b) s += gate_full[(size_t)b * NV * E_ + idx];
  gm[idx] = s * (1.0f / (float)B_);
}

// ---------------------------------------------------------------------------
// Kernel 3: mixture-of-experts GEMM on the f32 matrix cores.
//   D[p,n] = sum_e sum_s W_e[s,p] * (combine[n,e] * x[b,s,n])   + bias blend
// The top-2 combine weight is folded into the B-operand (per-lane scale), so
// all 8 experts chain into ONE v8f accumulator per 16x16 tile.
// Wave tile: 16p x 64n (4 n-subtiles). WG: 8 waves over p -> 128p x 64n.
// ---------------------------------------------------------------------------
__global__ __launch_bounds__(256) void moe_gemm_kernel(
    const float* __restrict__ X,    // [B,S,NV]
    const float* __restrict__ W,    // [E,S,P]
    const float* __restrict__ EB,   // [E,P]
    const float* __restrict__ C,    // combine [B,NV,E]
    float* __restrict__ out)        // [B,P,NV]
{
  const int lane = threadIdx.x;        // 0..31 (wave32)
  const int wave = threadIdx.y;        // 0..7
  const int lo   = lane & 15;
  const int hi   = lane >> 4;

  const int b      = blockIdx.x;
  const int n_base = blockIdx.y * 64;
  const int p_base = blockIdx.z * 128 + wave * 16;

  // Per-lane combine weights: lane 'lo' is column n of each B subtile.
  float ce[4][E_];
#pragma unroll
  for (int t = 0; t < 4; ++t) {
    const float* cp = C + ((size_t)b * NV + n_base + 16 * t + lo) * E_;
#pragma unroll
    for (int e = 0; e < E_; ++e) ce[t][e] = cp[e];
  }

  v8f acc[4];
#pragma unroll
  for (int t = 0; t < 4; ++t)
    acc[t] = (v8f){0.f, 0.f, 0.f, 0.f, 0.f, 0.f, 0.f, 0.f};

  const int    pA    = p_base + lo;        // A operand: lane row M -> p index
  const int    nB0   = n_base + lo;        // B operand: lane col N -> n index
  const size_t xbase = (size_t)b * S_ * NV;

  for (int s = 0; s < S_; s += 4) {
    const int r0 = s + 2 * hi;  // K-row for VGPR0 of A/B (lanes 16-31: K+2)

    // A tiles, one per expert: A[m=p, k] = W_e[s+k, p]  (p contiguous in mem)
    float a0[E_], a1[E_];
#pragma unroll
    for (int e = 0; e < E_; ++e) {
      const float* wp = W + ((size_t)e * S_ + r0) * P_ + pA;
      a0[e] = wp[0];
      a1[e] = wp[P_];
    }
    // x tiles, one per n-subtile: B[k, n] = x[b, s+k, n] (n contiguous in mem)
    float x0[4], x1[4];
#pragma unroll
    for (int t = 0; t < 4; ++t) {
      const float* xp = X + xbase + (size_t)r0 * NV + nB0 + 16 * t;
      x0[t] = xp[0];
      x1[t] = xp[NV];
    }

    if (s + 64 < S_) {  // speculative prefetch (global_prefetch_b8)
      __builtin_prefetch(X + xbase + (size_t)(s + 64 + 2 * hi) * NV + nB0, 0, 1);
      __builtin_prefetch(W + ((size_t)(s + 64 + 2 * hi)) * P_ + pA, 0, 1);
    }

#pragma unroll
    for (int e = 0; e < E_; ++e) {
      v2f A = {a0[e], a1[e]};
#pragma unroll
      for (int t = 0; t < 4; ++t) {
        v2f Bv = {ce[t][e] * x0[t], ce[t][e] * x1[t]};
        // D = A(16x4) x B(4x16) + C : v_wmma_f32_16x16x4_f32
        acc[t] = __builtin_amdgcn_wmma_f32_16x16x4_f32(
            false, A, false, Bv, (short)0, acc[t], false, false);
      }
    }
  }

  // Epilogue: blended bias + coalesced stores (lanes sweep n).
#pragma unroll
  for (int i = 0; i < 8; ++i) {
    const int p = p_base + i + 8 * hi;   // C/D layout: VGPR i -> M = i (+8 hi)
    float eb[E_];
#pragma unroll
    for (int e = 0; e < E_; ++e) eb[e] = EB[e * P_ + p];
#pragma unroll
    for (int t = 0; t < 4; ++t) {
      float bias = 0.0f;
#pragma unroll
      for (int e = 0; e < E_; ++e) bias = fmaf(ce[t][e], eb[e], bias);
      out[((size_t)b * P_ + p) * NV + n_base + 16 * t + lo] = acc[t][i] + bias;
    }
  }
}

// ---------------------------------------------------------------------------
extern "C" void kernel_launch(void* const* d_in, const int* in_sizes, int n_in,
                              void* d_out, int out_size, void* d_ws, size_t ws_size,
                              hipStream_t stream) {
  const float* x        = (const float*)d_in[0];  // [B,S,NV]
  const float* gate_W   = (const float*)d_in[1];  // [S,E]
  const float* expert_W = (const float*)d_in[2];  // [E,S,P]
  const float* expert_b = (const float*)d_in[3];  // [E,P]

  float* out       = (float*)d_out;                       // [B,P,NV]
  float* gate_mean = out + (size_t)B_ * P_ * NV;          // [NV,E]

  // Workspace: combine [B,NV,E] then gate_full [B,NV,E]  (2 MB total)
  float* combine   = (float*)d_ws;
  float* gate_full = combine + (size_t)B_ * NV * E_;

  gate_kernel<<<dim3((B_ * NV) / 256), dim3(256), 0, stream>>>(
      x, gate_W, combine, gate_full);
  gate_mean_kernel<<<dim3((NV * E_) / 256), dim3(256), 0, stream>>>(
      gate_full, gate_mean);
  moe_gemm_kernel<<<dim3(B_, NV / 64, P_ / 128), dim3(32, 8), 0, stream>>>(
      x, expert_W, expert_b, combine, out);
}